// DiceLossWithLongLabels_33655363732037
// MI455X (gfx1250) — compile-verified
//
#include <hip/hip_runtime.h>

typedef __attribute__((ext_vector_type(2))) float v2f;
typedef __attribute__((ext_vector_type(8))) float v8f;

#define NVOX  (128 * 128 * 128)
#define NCOL  80      // 0..34 inter(label-1), 40..74 union(label-1), padded to 5x16
#define UOFF  40
#define NBLK_MAX 1024

// ---------------------------------------------------------------------------
// Pass 1: per-voxel trilinear label gather + per-block LDS accumulation.
// Each block writes one fully-populated 80-float partial row to d_ws.
// ---------------------------------------------------------------------------
__global__ __launch_bounds__(256) void dice_accum_kernel(
    const int* __restrict__ ymov, const int* __restrict__ yfix,
    const float* __restrict__ warp, float* __restrict__ partial) {
  __shared__ float acc[NCOL];
  for (int i = threadIdx.x; i < NCOL; i += blockDim.x) acc[i] = 0.0f;
  __syncthreads();

  const int stride = gridDim.x * blockDim.x;
  for (int v = blockIdx.x * blockDim.x + threadIdx.x; v < NVOX; v += stride) {
    // warp layout [D,H,W,3] = (gx,gy,gz); unnormalize: ((g+1)*128-1)/2 = 64g+63.5
    const float gx = warp[3 * v + 0];
    const float gy = warp[3 * v + 1];
    const float gz = warp[3 * v + 2];
    const float ix = gx * 64.0f + 63.5f;
    const float iy = gy * 64.0f + 63.5f;
    const float iz = gz * 64.0f + 63.5f;
    const float fx0 = floorf(ix), fy0 = floorf(iy), fz0 = floorf(iz);
    const float fx = ix - fx0, fy = iy - fy0, fz = iz - fz0;
    const int x0 = (int)fx0, y0 = (int)fy0, z0 = (int)fz0;
    const float wx[2] = {1.0f - fx, fx};
    const float wy[2] = {1.0f - fy, fy};
    const float wz[2] = {1.0f - fz, fz};

    int   m[8];
    float w[8];
#pragma unroll
    for (int c = 0; c < 8; ++c) {
      const int dx = c & 1, dy = (c >> 1) & 1, dz = (c >> 2) & 1;
      const int xi = x0 + dx, yi = y0 + dy, zi = z0 + dz;
      const bool valid = ((unsigned)xi < 128u) & ((unsigned)yi < 128u) &
                         ((unsigned)zi < 128u);
      const int xc = min(max(xi, 0), 127);
      const int yc = min(max(yi, 0), 127);
      const int zc = min(max(zi, 0), 127);
      m[c] = ymov[(zc << 14) | (yc << 7) | xc];      // L2-resident gather
      w[c] = valid ? (wz[dz] * wy[dy] * wx[dx]) : 0.0f;
    }

    // fixed one-hot: inter and fixed^2 part of union
    const int f = yfix[v];
    if (f >= 1 && f <= 35) {
      float ia = 0.0f;
#pragma unroll
      for (int c = 0; c < 8; ++c) ia += (m[c] == f) ? w[c] : 0.0f;
      atomicAdd(&acc[f - 1], ia);
      atomicAdd(&acc[UOFF + f - 1], 1.0f);
    }

    // warped^2 part of union: one g^2 add per *distinct* corner label
#pragma unroll
    for (int c = 0; c < 8; ++c) {
      const int l = m[c];
      bool first = (l >= 1) && (l <= 35);
#pragma unroll
      for (int j = 0; j < c; ++j) first &= (m[j] != l);
      if (first) {
        float g = 0.0f;
#pragma unroll
        for (int j = 0; j < 8; ++j) g += (m[j] == l) ? w[j] : 0.0f;
        if (g != 0.0f) atomicAdd(&acc[UOFF + l - 1], g * g);
      }
    }
  }

  __syncthreads();
  for (int i = threadIdx.x; i < NCOL; i += blockDim.x)
    partial[blockIdx.x * NCOL + i] = acc[i];
}

// ---------------------------------------------------------------------------
// Pass 2: one wave32. Sum nblk x 80 partial rows with V_WMMA_F32_16X16X4_F32:
// D = ones(16x4) x B(4x16) + C  =>  every D row = column sums of 4 rows.
// Then finalize the 35 dice outputs.
// ---------------------------------------------------------------------------
__global__ __launch_bounds__(32) void dice_reduce_kernel(
    const float* __restrict__ partial, float* __restrict__ out, int nblk) {
  __shared__ float tot[NCOL];
  const int lane = threadIdx.x;          // 0..31
  const int krow = (lane >> 4) << 1;     // lanes 0-15 -> K=0/1, lanes 16-31 -> K=2/3
  const int col0 = lane & 15;            // N = lane & 15

  v2f a;
  a.x = 1.0f;
  a.y = 1.0f;

  for (int slab = 0; slab < 5; ++slab) {
    const int col = slab * 16 + col0;
    v8f c = {0.f, 0.f, 0.f, 0.f, 0.f, 0.f, 0.f, 0.f};
    for (int g = 0; g < nblk; g += 4) {
      v2f b;
      b.x = partial[(g + krow + 0) * NCOL + col];
      b.y = partial[(g + krow + 1) * NCOL + col];
      c = __builtin_amdgcn_wmma_f32_16x16x4_f32(
          /*neg_a=*/false, a, /*neg_b=*/false, b,
          /*c_mod=*/(short)0, c, /*reuse_a=*/false, /*reuse_b=*/false);
    }
    // All D rows identical; lane n and lane n+16 store the same value (benign).
    tot[col] = c[0];
  }
  __syncthreads();

  for (int l = lane; l < 35; l += 32) {
    const float inter = tot[l];
    const float uni = tot[UOFF + l];
    out[l] = 1.0f - (2.0f * inter + 1e-6f) / (uni + 1e-6f);
  }
}

// ---------------------------------------------------------------------------
extern "C" void kernel_launch(void* const* d_in, const int* in_sizes, int n_in,
                              void* d_out, int out_size, void* d_ws, size_t ws_size,
                              hipStream_t stream) {
  const int*   ymov = (const int*)d_in[0];    // [1,1,128,128,128] int32
  const int*   yfix = (const int*)d_in[1];    // [1,1,128,128,128] int32
  const float* warp = (const float*)d_in[2];  // [1,128,128,128,3] float32
  float* out = (float*)d_out;                 // [35] float32
  float* partial = (float*)d_ws;              // nblk x 80 floats

  int nblk = NBLK_MAX;
  const size_t rows_fit = ws_size / (NCOL * sizeof(float));
  if ((size_t)nblk > rows_fit) nblk = (int)(rows_fit & ~(size_t)3);
  if (nblk < 4) nblk = 4;

  dice_accum_kernel<<<nblk, 256, 0, stream>>>(ymov, yfix, warp, partial);
  dice_reduce_kernel<<<1, 32, 0, stream>>>(partial, out, nblk);
}